// GeometryRep_11450382811529
// MI455X (gfx1250) — compile-verified
//
#include <hip/hip_runtime.h>
#include <math.h>
#include <stdint.h>

// ---------------------------------------------------------------------------
// GeometryRep on MI455X (gfx1250, wave32):
//  * spatial-hash ball query (exact: radius <= cell size, top-10-in-radius
//    == masked global top-10)
//  * per-neighbor MLP 33->64->32->8 as f32 WMMA 16x16x4 tiles, one wave per
//    query (16 padded neighbor rows == one 16x16 D tile), mask-weighted
//    reduction via __shfl_xor across the wave32 halves.
//  * w2 weight preload via Tensor Data Mover (TENSOR_LOAD_TO_LDS +
//    s_wait_tensorcnt), with plain-load fallback.
// ---------------------------------------------------------------------------

#define NXg 64
#define NYg 32
#define NZg 32
#define MQ  65536      // NXg*NYg*NZg
#define R2c 0.0625f    // radius^2

typedef __attribute__((ext_vector_type(2))) float v2f;
typedef __attribute__((ext_vector_type(8))) float v8f;
typedef __attribute__((ext_vector_type(4))) unsigned int u32x4;
typedef __attribute__((ext_vector_type(8))) int i32x8;
typedef __attribute__((ext_vector_type(4))) int i32x4;

#if defined(__has_builtin)
#if __has_builtin(__builtin_amdgcn_tensor_load_to_lds) && \
    __has_builtin(__builtin_amdgcn_s_wait_tensorcnt)
#define GEO_HAVE_TDM 1
#endif
#endif

static __device__ __forceinline__ v8f wmma_f32_16x16x4(v2f a, v2f b, v8f c) {
  // 8 args: (neg_a, A, neg_b, B, c_mod, C, reuse_a, reuse_b)
  return __builtin_amdgcn_wmma_f32_16x16x4_f32(false, a, false, b, (short)0, c,
                                               false, false);
}

static __device__ __forceinline__ int clampi(int v, int lo, int hi) {
  return v < lo ? lo : (v > hi ? hi : v);
}

static __device__ __forceinline__ int cell_id(float x, float y, float z) {
  int cx = clampi((int)floorf((x + 1.0f) * 4.0f), 0, 7);
  int cy = clampi((int)floorf((y + 1.0f) * 4.0f), 0, 7);
  int cz = clampi((int)floorf((z + 1.0f) * 4.0f), 0, 7);
  return (cx * 8 + cy) * 8 + cz;
}

// ------------------------------ binning ------------------------------------

__global__ void geo_zero_i32(int* __restrict__ p, int n) {
  int i = blockIdx.x * blockDim.x + threadIdx.x;
  if (i < n) p[i] = 0;
}

__global__ void geo_count(const float* __restrict__ pts, int* __restrict__ cnt,
                          int n) {
  int i = blockIdx.x * blockDim.x + threadIdx.x;
  if (i >= n) return;
  atomicAdd(&cnt[cell_id(pts[i * 3 + 0], pts[i * 3 + 1], pts[i * 3 + 2])], 1);
}

__global__ void geo_scan(const int* __restrict__ cnt, int* __restrict__ start,
                         int* __restrict__ cursor) {
  __shared__ int tmp[512];
  int t = threadIdx.x;
  int c = cnt[t];
  tmp[t] = c;
  __syncthreads();
  for (int off = 1; off < 512; off <<= 1) {
    int v = (t >= off) ? tmp[t - off] : 0;
    __syncthreads();
    tmp[t] += v;
    __syncthreads();
  }
  int excl = tmp[t] - c;  // exclusive prefix
  start[t] = excl;
  cursor[t] = excl;
  if (t == 511) start[512] = tmp[511];
}

__global__ void geo_scatter(const float* __restrict__ pts,
                            int* __restrict__ cursor,
                            float* __restrict__ sorted, int n) {
  int i = blockIdx.x * blockDim.x + threadIdx.x;
  if (i >= n) return;
  float x = pts[i * 3 + 0], y = pts[i * 3 + 1], z = pts[i * 3 + 2];
  int pos = atomicAdd(&cursor[cell_id(x, y, z)], 1);
  sorted[pos * 3 + 0] = x;
  sorted[pos * 3 + 1] = y;
  sorted[pos * 3 + 2] = z;
}

// ------------------------------ ball query ---------------------------------
// One thread per query; 27-cell candidate scan (~2k candidates, L2-resident),
// fully-unrolled register insertion sort for top-10 within radius.

__global__ void geo_ball_topk(const float* __restrict__ grid,
                              const float* __restrict__ sorted,
                              const int* __restrict__ start,
                              float* __restrict__ nbr) {
  int q = blockIdx.x * blockDim.x + threadIdx.x;
  if (q >= MQ) return;
  float qx = grid[q * 3 + 0], qy = grid[q * 3 + 1], qz = grid[q * 3 + 2];

  float bestd[10];
  int besti[10];
#pragma unroll
  for (int k = 0; k < 10; ++k) {
    bestd[k] = 1e30f;
    besti[k] = -1;
  }

  int cx = clampi((int)floorf((qx + 1.0f) * 4.0f), 0, 7);
  int cy = clampi((int)floorf((qy + 1.0f) * 4.0f), 0, 7);
  int cz = clampi((int)floorf((qz + 1.0f) * 4.0f), 0, 7);

  for (int xx = (cx > 0 ? cx - 1 : 0); xx <= (cx < 7 ? cx + 1 : 7); ++xx)
    for (int yy = (cy > 0 ? cy - 1 : 0); yy <= (cy < 7 ? cy + 1 : 7); ++yy)
      for (int zz = (cz > 0 ? cz - 1 : 0); zz <= (cz < 7 ? cz + 1 : 7); ++zz) {
        int cell = (xx * 8 + yy) * 8 + zz;
        int s = start[cell], e = start[cell + 1];
        for (int i = s; i < e; ++i) {
          float dx = sorted[i * 3 + 0] - qx;
          float dy = sorted[i * 3 + 1] - qy;
          float dz = sorted[i * 3 + 2] - qz;
          float d2 = dx * dx + dy * dy + dz * dz;
          if (d2 <= R2c && d2 < bestd[9]) {
#pragma unroll
            for (int k = 9; k >= 0; --k) {
              if (d2 < bestd[k]) {
                if (k > 0 && d2 < bestd[k - 1]) {
                  bestd[k] = bestd[k - 1];
                  besti[k] = besti[k - 1];
                } else {
                  bestd[k] = d2;
                  besti[k] = i;
                }
              }
            }
          }
        }
      }

#pragma unroll
  for (int k = 0; k < 16; ++k) {
    float ox = 0.0f, oy = 0.0f, oz = 0.0f;
    if (k < 10 && besti[k] >= 0) {
      ox = sorted[besti[k] * 3 + 0];
      oy = sorted[besti[k] * 3 + 1];
      oz = sorted[besti[k] * 3 + 2];
    }
    size_t b = ((size_t)q * 16 + k) * 3;
    nbr[b + 0] = ox;
    nbr[b + 1] = oy;
    nbr[b + 2] = oz;
  }
}

// ------------------------------ WMMA MLP -----------------------------------
// A fragment (16x4 f32): lane<16 holds (K=4c,4c+1) of row lane; lane>=16 holds
// (K=4c+2,4c+3) of row lane-16. B fragment mirrored (rows K striped across
// halves, col = lane%16). C/D: VGPR v, lane L -> D[v + 8*(L/16)][L%16].

__device__ __forceinline__ void geo_layer_gelu(
    const float* __restrict__ A, int as, const float* __restrict__ W,
    const float* __restrict__ B, float* __restrict__ H, int hs, int KD, int OD,
    int half, int r) {
  for (int nt = 0; nt < OD / 16; ++nt) {
    v8f acc;
    float bv = B[nt * 16 + r];
#pragma unroll
    for (int i = 0; i < 8; ++i) acc[i] = bv;
    for (int c = 0; c < KD / 4; ++c) {
      int k0 = 4 * c + 2 * half;
      v2f av, bw;
      av.x = A[r * as + k0];
      av.y = A[r * as + k0 + 1];
      bw.x = W[k0 * OD + nt * 16 + r];
      bw.y = W[(k0 + 1) * OD + nt * 16 + r];
      acc = wmma_f32_16x16x4(av, bw, acc);
    }
#pragma unroll
    for (int v = 0; v < 8; ++v) {
      float x = acc[v];
      x = 0.5f * x * (1.0f + erff(x * 0.70710678118654752f));  // exact GELU
      H[(v + 8 * half) * hs + nt * 16 + r] = x;
    }
  }
}

__global__ void __launch_bounds__(128) geo_mlp_wmma(
    const float* __restrict__ nbr, const float* __restrict__ w1,
    const float* __restrict__ b1, const float* __restrict__ w2,
    const float* __restrict__ b2, const float* __restrict__ w3,
    const float* __restrict__ b3, const float* __restrict__ freqs,
    float* __restrict__ out) {
  __shared__ float w1s[36 * 64];  // 33 rows real, 3 zero-pad
  __shared__ float w2s[64 * 32];
  __shared__ float w3s[32 * 16];  // 8 cols real, 8 zero-pad
  __shared__ float b1s[64], b2s[32], b3s[16];
  __shared__ float featS[4][16 * 36];  // stride 36 -> distinct LDS banks
  __shared__ float h1S[4][16 * 68];    // stride 68 -> distinct LDS banks
  __shared__ float h2S[4][16 * 36];
  __shared__ float maskS[4][16];

  int tid = threadIdx.x;

#ifdef GEO_HAVE_TDM
  // Wave 0 issues a TDM flat 1-D tile copy of w2 (64x32 f32 = 2048 elements)
  // straight into LDS; TENSORcnt tracks completion.
  if (tid < 32) {
    unsigned long long ga = (unsigned long long)(uintptr_t)w2;
    unsigned int la = (unsigned int)(uintptr_t)(void*)w2s;
    u32x4 g0 = {1u,  // count=1, is_restore=0, gather off
                la,  // lds_addr (bytes)
                (unsigned int)(ga & 0xffffffffu),
                (unsigned int)((ga >> 32) & 0x1ffffffu) | (2u << 30)};  // type=2
    i32x8 g1 = {(int)(2u << 16),     // data_size = 4B; workgroup_mask = 0
                (int)(2048u << 16),  // tensor_dim0 = 2048 (bits 63:48)
                (int)(1u << 16),     // tensor_dim1 = 1    (bits 95:80)
                (int)(2048u << 16),  // tile_dim0  = 2048  (bits 127:112)
                0,                   // tile_dim1/2 unused
                2048,                // tensor_dim0_stride lo
                0, 0};
    i32x4 gz = {0, 0, 0, 0};
#if __clang_major__ >= 23
    i32x8 gz8 = {0, 0, 0, 0, 0, 0, 0, 0};
    __builtin_amdgcn_tensor_load_to_lds(g0, g1, gz, gz, gz8, 0);
#else
    __builtin_amdgcn_tensor_load_to_lds(g0, g1, gz, gz, 0);
#endif
    __builtin_amdgcn_s_wait_tensorcnt(0);
  }
#else
  for (int i = tid; i < 64 * 32; i += 128) w2s[i] = w2[i];
#endif

  for (int i = tid; i < 36 * 64; i += 128) {
    int rr = i / 64, cc = i % 64;
    w1s[i] = (rr < 33) ? w1[rr * 64 + cc] : 0.0f;
  }
  for (int i = tid; i < 32 * 16; i += 128) {
    int rr = i / 16, cc = i % 16;
    w3s[i] = (cc < 8) ? w3[rr * 8 + cc] : 0.0f;
  }
  if (tid < 64) b1s[tid] = b1[tid];
  if (tid < 32) b2s[tid] = b2[tid];
  if (tid < 16) b3s[tid] = (tid < 8) ? b3[tid] : 0.0f;
  __syncthreads();

  int wave = tid >> 5, lane = tid & 31;
  int half = lane >> 4, r = lane & 15;
  int q = blockIdx.x * 4 + wave;  // one wave == one query (16 padded rows)
  float* feat = featS[wave];
  float* h1v = h1S[wave];
  float* h2v = h2S[wave];

  // features: [x,y,z, sin(d*f) f-major, cos(d*f) f-major, 0,0,0]
  if (lane < 16) {
    int row = lane;
    size_t base = ((size_t)q * 16 + row) * 3;
    float x = nbr[base + 0], y = nbr[base + 1], z = nbr[base + 2];
    feat[row * 36 + 0] = x;
    feat[row * 36 + 1] = y;
    feat[row * 36 + 2] = z;
#pragma unroll
    for (int f = 0; f < 5; ++f) {
      float fr = freqs[f];
      float s0, c0, s1, c1, s2, c2;
      sincosf(x * fr, &s0, &c0);
      sincosf(y * fr, &s1, &c1);
      sincosf(z * fr, &s2, &c2);
      feat[row * 36 + 3 + f * 3 + 0] = s0;
      feat[row * 36 + 3 + f * 3 + 1] = s1;
      feat[row * 36 + 3 + f * 3 + 2] = s2;
      feat[row * 36 + 18 + f * 3 + 0] = c0;
      feat[row * 36 + 18 + f * 3 + 1] = c1;
      feat[row * 36 + 18 + f * 3 + 2] = c2;
    }
    feat[row * 36 + 33] = 0.0f;
    feat[row * 36 + 34] = 0.0f;
    feat[row * 36 + 35] = 0.0f;
    maskS[wave][row] = (fabsf(x) > 1e-6f) ? 1.0f : 0.0f;
  }
  __syncthreads();

  geo_layer_gelu(feat, 36, w1s, b1s, h1v, 68, 36, 64, half, r);  // 33->64
  __syncthreads();
  geo_layer_gelu(h1v, 68, w2s, b2s, h2v, 36, 64, 32, half, r);   // 64->32
  __syncthreads();

  // layer 3 (32->8, tanh) + mask-weighted sum over the 16 neighbor rows
  {
    v8f acc;
    float bv = b3s[r];
#pragma unroll
    for (int i = 0; i < 8; ++i) acc[i] = bv;
#pragma unroll
    for (int c = 0; c < 8; ++c) {
      int k0 = 4 * c + 2 * half;
      v2f av, bw;
      av.x = h2v[r * 36 + k0];
      av.y = h2v[r * 36 + k0 + 1];
      bw.x = w3s[k0 * 16 + r];
      bw.y = w3s[(k0 + 1) * 16 + r];
      acc = wmma_f32_16x16x4(av, bw, acc);
    }
    float s = 0.0f;
#pragma unroll
    for (int v = 0; v < 8; ++v)
      s += tanhf(acc[v]) * maskS[wave][v + 8 * half];
    s += __shfl_xor(s, 16, 32);  // combine row halves (wave32)
    if (lane < 8) out[(size_t)lane * MQ + q] = s;  // (8, NX, NY, NZ)
  }
}

// ------------------------------ launch -------------------------------------

extern "C" void kernel_launch(void* const* d_in, const int* in_sizes, int n_in,
                              void* d_out, int out_size, void* d_ws,
                              size_t ws_size, hipStream_t stream) {
  const float* pts = (const float*)d_in[0];
  const float* grid = (const float*)d_in[1];
  const float* w1 = (const float*)d_in[2];
  const float* b1 = (const float*)d_in[3];
  const float* w2 = (const float*)d_in[4];
  const float* b2 = (const float*)d_in[5];
  const float* w3 = (const float*)d_in[6];
  const float* b3 = (const float*)d_in[7];
  const float* freqs = (const float*)d_in[8];
  float* out = (float*)d_out;
  int npts = in_sizes[0] / 3;

  char* ws = (char*)d_ws;
  int* cnt = (int*)ws;                              // 512 ints
  int* start = (int*)(ws + 4096);                   // 513 ints
  int* cursor = (int*)(ws + 4096 + 4096);           // 512 ints
  float* sorted = (float*)(ws + 3 * 4096);          // npts*3 floats
  size_t sortedBytes = (((size_t)npts * 3 * sizeof(float)) + 255) & ~(size_t)255;
  float* nbr = (float*)(ws + 3 * 4096 + sortedBytes);  // MQ*16*3 floats

  geo_zero_i32<<<2, 256, 0, stream>>>(cnt, 512);
  geo_count<<<(npts + 255) / 256, 256, 0, stream>>>(pts, cnt, npts);
  geo_scan<<<1, 512, 0, stream>>>(cnt, start, cursor);
  geo_scatter<<<(npts + 255) / 256, 256, 0, stream>>>(pts, cursor, sorted,
                                                      npts);
  geo_ball_topk<<<MQ / 256, 256, 0, stream>>>(grid, sorted, start, nbr);
  geo_mlp_wmma<<<MQ / 4, 128, 0, stream>>>(nbr, w1, b1, w2, b2, w3, b3, freqs,
                                           out);
}